// MemoryBank_43980465111532
// MI455X (gfx1250) — compile-verified
//
#include <hip/hip_runtime.h>
#include <math.h>

typedef __attribute__((ext_vector_type(2))) float v2f;
typedef __attribute__((ext_vector_type(8))) float v8f;

#define N_DIM 256
#define Q_DIM 64
#define EPS_F 1e-9f
#define SENTINEL 0xFFFFFFFFFFFFFFFFull

// Monotone map: f32 -> u32 such that (a < b) <=> (key(a) < key(b)).
__device__ __forceinline__ unsigned int fkey(float v) {
    unsigned int u = __float_as_uint(v);
    return (u & 0x80000000u) ? ~u : (u | 0x80000000u);
}

__global__ void mb_init_table(unsigned long long* __restrict__ tab, int n) {
    int i = blockIdx.x * blockDim.x + threadIdx.x;
    if (i < n) tab[i] = SENTINEL;
}

// Phase 1: one block per detection b; 4 waves; wave w computes sim[b, 16w .. 16w+15]
// via 64 chained V_WMMA_F32_16X16X4_F32, then publishes segment-argmin with a
// packed (value,index) u64 atomicMin (value-min, then lowest-b tie-break).
__global__ void __launch_bounds__(128)
mb_sim_argmin(const float* __restrict__ reprs,
              const float* __restrict__ memory,
              const int*   __restrict__ tid,
              unsigned long long* __restrict__ tab) {
    const int b    = blockIdx.x;
    const int lane = threadIdx.x & 31;
    const int wave = threadIdx.x >> 5;
    const int q0   = wave * 16;
    const int t    = tid[b];

    // A-matrix (16x4 f32) per-lane layout: lane L holds (M = L%16, K = 2*(L/16) + {0,1}).
    // B-matrix (4x16 f32) rows replicated across columns -> same per-lane pair indexing.
    const int half = lane >> 4;           // 0 or 1
    const int m    = lane & 15;           // row within q-tile
    const float* __restrict__ arow = memory + ((size_t)t * Q_DIM + q0 + m) * N_DIM + 2 * half;
    const float* __restrict__ brow = reprs  + (size_t)b * N_DIM + 2 * half;

    v8f c = {};
#pragma unroll 4
    for (int k = 0; k < N_DIM; k += 4) {
        v2f av = *(const v2f*)(arow + k);
        v2f bv = *(const v2f*)(brow + k);
        // 8 args: (neg_a, A, neg_b, B, c_mod, C, reuse_a, reuse_b)
        c = __builtin_amdgcn_wmma_f32_16x16x4_f32(false, av, false, bv,
                                                  (short)0, c, false, false);
    }

    // D layout (16x16 f32): VGPR r = row r (lanes 0-15) / row 8+r (lanes 16-31);
    // every column identical, so lanes 0 and 16 publish 8 rows each.
    if (m == 0) {
        unsigned long long* base = tab + (size_t)t * Q_DIM + q0 + 8 * half;
#pragma unroll
        for (int r = 0; r < 8; ++r) {
            unsigned long long key =
                ((unsigned long long)fkey(c[r]) << 32) | (unsigned int)b;
            atomicMin(base + r, key);
        }
    }
}

// Phase 3: one wave per (t,q) row of 256 floats. Blend + renormalize, or copy-through
// for tracks with no detections.
__global__ void __launch_bounds__(256)
mb_update(const float* __restrict__ reprs,
          const float* __restrict__ memory,
          const float* __restrict__ alpha,
          const unsigned long long* __restrict__ tab,
          float* __restrict__ out, int TQ) {
    const int lane = threadIdx.x & 31;
    const int wave = threadIdx.x >> 5;
    const int row  = blockIdx.x * (blockDim.x >> 5) + wave;
    if (row >= TQ) return;

    const float* __restrict__ mrow = memory + (size_t)row * N_DIM;
    float* __restrict__ orow = out + (size_t)row * N_DIM;
    const unsigned long long p = tab[row];

    if (p == SENTINEL) {  // track absent: pass memory through
#pragma unroll
        for (int j = 0; j < N_DIM / 32; ++j)
            orow[lane + 32 * j] = mrow[lane + 32 * j];
        return;
    }

    const int q     = row & (Q_DIM - 1);
    const int bstar = (int)(p & 0xFFFFFFFFull);
    const float* __restrict__ rrow = reprs + (size_t)bstar * N_DIM;
    const float a = alpha[q], oma = 1.0f - a;

    float v[N_DIM / 32];
    float ss = 0.0f;
#pragma unroll
    for (int j = 0; j < N_DIM / 32; ++j) {
        const float mv = mrow[lane + 32 * j];
        const float rv = rrow[lane + 32 * j];
        v[j] = a * mv + oma * rv;
        ss  += v[j] * v[j];
    }
#pragma unroll
    for (int off = 16; off > 0; off >>= 1)
        ss += __shfl_xor(ss, off, 32);     // wave32 butterfly reduction

    const float inv = 1.0f / (sqrtf(ss) + EPS_F);
#pragma unroll
    for (int j = 0; j < N_DIM / 32; ++j)
        orow[lane + 32 * j] = v[j] * inv;
}

extern "C" void kernel_launch(void* const* d_in, const int* in_sizes, int n_in,
                              void* d_out, int out_size, void* d_ws, size_t ws_size,
                              hipStream_t stream) {
    const float* reprs  = (const float*)d_in[0];   // [B, 256]
    const float* memory = (const float*)d_in[1];   // [T, 64, 256]
    const float* alpha  = (const float*)d_in[2];   // [64]
    const int*   tid    = (const int*)d_in[3];     // [B]
    float* out = (float*)d_out;                    // [T, 64, 256]

    const int B  = in_sizes[0] / N_DIM;
    const int TQ = in_sizes[1] / N_DIM;            // T * Q

    unsigned long long* tab = (unsigned long long*)d_ws;  // T*Q * 8B = 2 MB

    mb_init_table<<<(TQ + 255) / 256, 256, 0, stream>>>(tab, TQ);
    mb_sim_argmin<<<B, 128, 0, stream>>>(reprs, memory, tid, tab);

    const int wavesPerBlock = 8;
    mb_update<<<(TQ + wavesPerBlock - 1) / wavesPerBlock, wavesPerBlock * 32, 0, stream>>>(
        reprs, memory, alpha, tab, out, TQ);
}